// Caps1D_56315611185422
// MI455X (gfx1250) — compile-verified
//
#include <hip/hip_runtime.h>
#include <math.h>

// CapsNet dynamic routing, fully fused, CDNA5 (gfx1250) wave32 + V_WMMA_F32_16X16X4_F32.
// One workgroup = 16 batch rows x one class k. Routing logits live in LDS; u_ji is
// recomputed from u/W on every pass via chained f32 WMMAs (never materialized).
// Wave-private staging (DS in-order per wave) -> no workgroup barriers in hot loops.
// Software-pipelined: next chunk's global loads issue before current chunk's WMMAs.

typedef __attribute__((ext_vector_type(2))) float v2f;
typedef __attribute__((ext_vector_type(8))) float v8f;

#define R_RT   2336
#define M_IN   4
#define P_OUT  16
#define K_CLS  2
#define BT     16            // batch tile (== WMMA M)
#define NCHUNK (R_RT / 16)   // 146: 16 routes consumed per workgroup per step
#define NTH    256           // 8 wave32

// per-wave staging slice sizes (floats)
#define A_U_STRIDE 12        // pass A u: [bi][rr*4+m], 16*12 = 192
#define A_SLICE    328       // 192 (u) + 136 (W: 2 routes * 68)
#define B_U_STRIDE 20        // pass B u: [bi][rr*4+m], 16*20 = 320
#define B_SLICE    592       // 320 (u) + 272 (W: 4 routes * 68)
#define STG_FLOATS 2624      // max(8*A_SLICE, 4*B_SLICE)

__global__ __launch_bounds__(NTH, 1)
void caps_route_kernel(const float* __restrict__ U,   // [B, R, M]
                       const float* __restrict__ Wt,  // [K, R, M, P]
                       float* __restrict__ Out)       // [B, K]
{
    // ---- LDS: 162,240 bytes ----
    __shared__ float bl[R_RT * BT];     // routing logits [r][bi]            149504 B
    __shared__ float stg[STG_FLOATS];   // per-wave staging pool              10496 B
    __shared__ float sAcc[256];         // s accumulator [bi][p] / softmax scratch
    __shared__ float vS[BT * 17];       // squashed v [bi][p], stride 17
    __shared__ float mxS[BT];
    __shared__ float izS[BT];

    const int k    = blockIdx.y;
    const int b0   = blockIdx.x * BT;
    const int t    = threadIdx.x;
    const int wave = t >> 5;
    const int lane = t & 31;
    const int li   = lane & 15;   // WMMA row/col role
    const int hi   = lane >> 4;   // lane half

    const float* Wk = Wt + (size_t)k * R_RT * M_IN * P_OUT;

    for (int i = t; i < R_RT * BT; i += NTH) bl[i] = 0.0f;

    // pass-A staging roles (wave-private): u covers 16 bi x 2 rr, W covers 2 rr x 16 f4
    const int biu  = lane >> 1, rru = lane & 1;       // u role
    const int mW   = (lane >> 2) & 3, pqW = lane & 3; // W role (rr_w == hi)
    float* u2 = &stg[wave * A_SLICE];
    float* w2 = &stg[wave * A_SLICE + 192];
    const float* uA = U  + ((size_t)(b0 + biu) * R_RT + rru) * M_IN;       // + r*4
    const float* wA = Wk + ((size_t)hi * 64 + mW * 16 + pqW * 4);          // + r*64

    for (int it = 0; it < 3; ++it) {
        float mx = 0.0f, iz = 0.0f;
        // ---------- softmax stats over r per batch row (iter 0: uniform) ----------
        if (it > 0) {
            __syncthreads();                     // pass-B bl writes complete
            const int bi = t & 15, sl = t >> 4;  // 16 slices x 16 bi
            float m = -3.4e38f;
            for (int r = sl; r < R_RT; r += 16) m = fmaxf(m, bl[r * BT + bi]);
            sAcc[sl * 16 + bi] = m;
            __syncthreads();
            if (t < BT) {
                float mm = sAcc[t];
                for (int s = 1; s < 16; ++s) mm = fmaxf(mm, sAcc[s * 16 + t]);
                mxS[t] = mm;
            }
            __syncthreads();
            float z = 0.0f;
            for (int r = sl; r < R_RT; r += 16) z += __expf(bl[r * BT + bi] - mxS[bi]);
            sAcc[sl * 16 + bi] = z;
            __syncthreads();
            if (t < BT) {
                float zz = 0.0f;
                for (int s = 0; s < 16; ++s) zz += sAcc[s * 16 + t];
                izS[t] = 1.0f / zz;
            }
            __syncthreads();
            mx = mxS[li];                        // hoisted out of the hot loop
            iz = izS[li];
        }
        __syncthreads();
        sAcc[t] = 0.0f;                          // NTH == 256 == sAcc size
        __syncthreads();

        // ---------- pass A: s[bi,p] = sum_{r,m} (c*u)[bi,(r,m)] * W[(r,m),p] ----------
        // Each wave owns routes r = ch*16 + wave*2 + {0,1}; chained k=4 f32 WMMAs.
        // Pipelined: stage ch from regs, issue loads for ch+1, then consume ch.
        v8f acc = {};
        const float cu = 1.0f / (float)R_RT;
        float4 uvR = *(const float4*)(uA + (size_t)(wave * 2) * M_IN);
        float4 wvR = *(const float4*)(wA + (size_t)(wave * 2) * 64);

        #define PASSA_STAGE()                                                        \
            *(float4*)(&u2[biu * A_U_STRIDE + rru * 4]) = uvR;                       \
            w2[hi * 68 + (pqW * 4 + 0) * 4 + mW] = wvR.x;                            \
            w2[hi * 68 + (pqW * 4 + 1) * 4 + mW] = wvR.y;                            \
            w2[hi * 68 + (pqW * 4 + 2) * 4 + mW] = wvR.z;                            \
            w2[hi * 68 + (pqW * 4 + 3) * 4 + mW] = wvR.w;                            \
            if (ch + 1 < NCHUNK) {                                                   \
                uvR = *(const float4*)(uA + (size_t)(r0 + 16) * M_IN);               \
                wvR = *(const float4*)(wA + (size_t)(r0 + 16) * 64);                 \
                if (ch + 2 < NCHUNK)                                                 \
                    __builtin_prefetch((const void*)(uA + (size_t)(r0 + 32) * M_IN), 0, 1); \
            }

        if (it == 0) {
            for (int ch = 0; ch < NCHUNK; ++ch) {
                const int r0 = ch * 16 + wave * 2;
                PASSA_STAGE()
                __builtin_amdgcn_wave_barrier();
                #pragma unroll
                for (int j = 0; j < 2; ++j) {
                    v2f a, b;
                    a.x = cu * u2[li * A_U_STRIDE + j * 4 + 2 * hi + 0];
                    a.y = cu * u2[li * A_U_STRIDE + j * 4 + 2 * hi + 1];
                    b.x = w2[j * 68 + li * 4 + 2 * hi + 0];
                    b.y = w2[j * 68 + li * 4 + 2 * hi + 1];
                    acc = __builtin_amdgcn_wmma_f32_16x16x4_f32(
                              false, a, false, b, (short)0, acc, false, false);
                }
                __builtin_amdgcn_wave_barrier();
            }
        } else {
            for (int ch = 0; ch < NCHUNK; ++ch) {
                const int r0 = ch * 16 + wave * 2;
                PASSA_STAGE()
                __builtin_amdgcn_wave_barrier();
                #pragma unroll
                for (int j = 0; j < 2; ++j) {
                    const float cb = __expf(bl[(r0 + j) * BT + li] - mx) * iz;
                    v2f a, b;
                    a.x = cb * u2[li * A_U_STRIDE + j * 4 + 2 * hi + 0];
                    a.y = cb * u2[li * A_U_STRIDE + j * 4 + 2 * hi + 1];
                    b.x = w2[j * 68 + li * 4 + 2 * hi + 0];
                    b.y = w2[j * 68 + li * 4 + 2 * hi + 1];
                    acc = __builtin_amdgcn_wmma_f32_16x16x4_f32(
                              false, a, false, b, (short)0, acc, false, false);
                }
                __builtin_amdgcn_wave_barrier();
            }
        }
        #undef PASSA_STAGE

        // cross-wave reduce: D VGPR j -> row (j + 8*hi) = batch, col li = p
        for (int j2 = 0; j2 < 8; ++j2)
            atomicAdd(&sAcc[(j2 + 8 * hi) * 16 + li], acc[j2]);
        __syncthreads();

        // ---------- squash; final iteration emits |v| = n/(1+n) ----------
        if (t < BT) {
            float norm = 0.0f;
            for (int p = 0; p < P_OUT; ++p) { float s = sAcc[t * 16 + p]; norm += s * s; }
            const float f = sqrtf(norm) / (1.0f + norm);
            for (int p = 0; p < P_OUT; ++p) vS[t * 17 + p] = sAcc[t * 16 + p] * f;
            if (it == 2) Out[(size_t)(b0 + t) * K_CLS + k] = norm / (1.0f + norm);
        }
        __syncthreads();
        if (it == 2) break;

        // ---------- pass B: bl[bi,r] += sum_m u[bi,r,m] * (sum_p W[(r,m),p] v[bi,p]) ----------
        // Waves 0-3 each own 4 routes (16 rm rows) per step; pipelined wave-private staging.
        if (wave < 4) {
            float* u4 = &stg[wave * B_SLICE];
            float* w4 = &stg[wave * B_SLICE + 320];
            const int rrA = li >> 2, mA = li & 3;     // A row = (r,m) flat
            const int bu0 = lane >> 2,        ru0 = lane & 3;          // u role, i=0/1
            const int rw0 = lane >> 4,        mw0 = (lane >> 2) & 3, pw0 = lane & 3;
            const float* uB = U  + ((size_t)(b0 + bu0) * R_RT + ru0) * M_IN;  // + r*4
            const float* wB0 = Wk + ((size_t)rw0       * 64 + mw0 * 16 + pw0 * 4);
            const float* wB1 = Wk + ((size_t)(rw0 + 2) * 64 + mw0 * 16 + pw0 * 4);
            // i=1 covers bi rows 8..15 and W routes rw0+2 (idx = lane+32)
            const float* uB1 = U + ((size_t)(b0 + bu0 + 8) * R_RT + ru0) * M_IN;

            float4 ur0 = *(const float4*)(uB  + (size_t)(wave * 4) * M_IN);
            float4 ur1 = *(const float4*)(uB1 + (size_t)(wave * 4) * M_IN);
            float4 wr0 = *(const float4*)(wB0 + (size_t)(wave * 4) * 64);
            float4 wr1 = *(const float4*)(wB1 + (size_t)(wave * 4) * 64);

            for (int ch = 0; ch < NCHUNK; ++ch) {
                const int r0 = ch * 16 + wave * 4;
                *(float4*)(&u4[bu0 * B_U_STRIDE + ru0 * 4]) = ur0;
                *(float4*)(&u4[(bu0 + 8) * B_U_STRIDE + ru0 * 4]) = ur1;
                w4[rw0 * 68 + (pw0 * 4 + 0) * 4 + mw0] = wr0.x;
                w4[rw0 * 68 + (pw0 * 4 + 1) * 4 + mw0] = wr0.y;
                w4[rw0 * 68 + (pw0 * 4 + 2) * 4 + mw0] = wr0.z;
                w4[rw0 * 68 + (pw0 * 4 + 3) * 4 + mw0] = wr0.w;
                w4[(rw0 + 2) * 68 + (pw0 * 4 + 0) * 4 + mw0] = wr1.x;
                w4[(rw0 + 2) * 68 + (pw0 * 4 + 1) * 4 + mw0] = wr1.y;
                w4[(rw0 + 2) * 68 + (pw0 * 4 + 2) * 4 + mw0] = wr1.z;
                w4[(rw0 + 2) * 68 + (pw0 * 4 + 3) * 4 + mw0] = wr1.w;
                if (ch + 1 < NCHUNK) {
                    ur0 = *(const float4*)(uB  + (size_t)(r0 + 16) * M_IN);
                    ur1 = *(const float4*)(uB1 + (size_t)(r0 + 16) * M_IN);
                    wr0 = *(const float4*)(wB0 + (size_t)(r0 + 16) * 64);
                    wr1 = *(const float4*)(wB1 + (size_t)(r0 + 16) * 64);
                }
                __builtin_amdgcn_wave_barrier();
                v8f g = {};
                #pragma unroll
                for (int kk = 0; kk < 4; ++kk) {      // reduce over p in 4 k=4 steps
                    const int p0 = kk * 4 + 2 * hi;
                    v2f a, b;
                    a.x = w4[rrA * 68 + (p0 + 0) * 4 + mA];  // A = W_flat [16 rm x 4 p]
                    a.y = w4[rrA * 68 + (p0 + 1) * 4 + mA];
                    b.x = vS[li * 17 + p0 + 0];              // B = v^T [4 p x 16 bi]
                    b.y = vS[li * 17 + p0 + 1];
                    g = __builtin_amdgcn_wmma_f32_16x16x4_f32(
                            false, a, false, b, (short)0, g, false, false);
                }
                // D[rm = j2+8*hi, bi = li]; contract with u over m, update logits
                float a0 = 0.0f, a1 = 0.0f;
                #pragma unroll
                for (int j2 = 0; j2 < 8; ++j2) {
                    const int rm = 8 * hi + j2;
                    const float uvv = u4[li * B_U_STRIDE + (rm >> 2) * 4 + (rm & 3)];
                    if (j2 < 4) a0 += uvv * g[j2]; else a1 += uvv * g[j2];
                }
                const int rg = r0 + 2 * hi;               // unique owner per (bi, r)
                bl[(rg + 0) * BT + li] += a0;
                bl[(rg + 1) * BT + li] += a1;
                __builtin_amdgcn_wave_barrier();
            }
        }
    }
}

extern "C" void kernel_launch(void* const* d_in, const int* in_sizes, int n_in,
                              void* d_out, int out_size, void* d_ws, size_t ws_size,
                              hipStream_t stream) {
    const float* u = (const float*)d_in[0];   // [1024, 2336, 4] f32
    const float* W = (const float*)d_in[1];   // [2, 2336, 4, 16] f32
    float* out = (float*)d_out;               // [1024, 2] f32
    const int Btotal = in_sizes[0] / (R_RT * M_IN);   // 1024
    dim3 grid(Btotal / BT, K_CLS, 1);
    caps_route_kernel<<<grid, NTH, 0, stream>>>(u, W, out);
}